// GAT_1872605741067
// MI455X (gfx1250) — compile-verified
//
#include <hip/hip_runtime.h>
#include <math.h>

#define HEADS 4
#define DH    64
#define F     256      // = HEADS*DH = IN
#define SLOPE 0.2f

typedef float v2f __attribute__((ext_vector_type(2)));
typedef float v8f __attribute__((ext_vector_type(8)));

// ---------------------------------------------------------------------------
// small int utility kernels (CSR build)
// ---------------------------------------------------------------------------
__global__ void fill_int_kernel(int* __restrict__ p, int v, int n) {
  int i = blockIdx.x * blockDim.x + threadIdx.x;
  int stride = gridDim.x * blockDim.x;
  for (; i < n; i += stride) p[i] = v;
}

__global__ void copy_int_kernel(int* __restrict__ dstp, const int* __restrict__ srcp, int n) {
  int i = blockIdx.x * blockDim.x + threadIdx.x;
  int stride = gridDim.x * blockDim.x;
  for (; i < n; i += stride) dstp[i] = srcp[i];
}

__global__ void count_kernel(const int* __restrict__ dst, int* __restrict__ deg, int Ee) {
  int e = blockIdx.x * blockDim.x + threadIdx.x;
  if (e < Ee) atomicAdd(&deg[dst[e]], 1);
}

// single-block chunked exclusive scan (N=50000 -> trivial runtime)
__global__ void scan_kernel(const int* __restrict__ deg, int* __restrict__ off, int Nn) {
  __shared__ int lds[256];
  __shared__ int carry_s;
  int t = threadIdx.x;
  if (t == 0) carry_s = 0;
  __syncthreads();
  for (int base = 0; base < Nn; base += 256) {
    int i = base + t;
    int v = (i < Nn) ? deg[i] : 0;
    lds[t] = v;
    __syncthreads();
    #pragma unroll
    for (int ofs = 1; ofs < 256; ofs <<= 1) {
      int add = (t >= ofs) ? lds[t - ofs] : 0;
      __syncthreads();
      lds[t] += add;
      __syncthreads();
    }
    int incl = lds[t];
    int carry = carry_s;
    if (i < Nn) off[i] = carry + incl - v;   // exclusive
    __syncthreads();
    if (t == 255) carry_s = carry + incl;
    __syncthreads();
  }
  if (t == 0) off[Nn] = carry_s;
}

__global__ void scatter_kernel(const int* __restrict__ dst, int* __restrict__ cursor,
                               int* __restrict__ eids, int Ee) {
  int e = blockIdx.x * blockDim.x + threadIdx.x;
  if (e >= Ee) return;
  int pos = atomicAdd(&cursor[dst[e]], 1);
  eids[pos] = e;
}

// ---------------------------------------------------------------------------
// C[M,256] = A[M,256] @ B[256,256], f32 row-major, exact f32 math.
// One wave per 16(M)x64(N) strip: 4 independent accumulators -> consecutive
// V_WMMA_F32_16X16X4_F32 ops have no D->C dependency chain; A fragment load
// amortized over 4 WMMAs.
// A 16x4: lane l holds row (l&15); VGPR v holds K=(l>>4)*2+v.
// B 4x16: lane l holds col (l&15); same K striping.
// C/D:    VGPR v -> row v+(l>>4)*8, col (l&15).
// ---------------------------------------------------------------------------
__global__ void gemm256_wmma(const float* __restrict__ A, const float* __restrict__ B,
                             float* __restrict__ C, int M) {
  const int K = F, NC = F;
  int gwave = (int)((blockIdx.x * blockDim.x + threadIdx.x) >> 5);
  int lane  = threadIdx.x & 31;
  const int ntn = NC / 64;                 // 4 column strips
  int tile_m = gwave / ntn;
  int tile_n = gwave % ntn;
  if (tile_m * 16 >= M) return;            // wave-uniform -> EXEC stays full

  int lm = lane & 15;
  int kk = (lane >> 4) << 1;               // 0 or 2
  const float* Arow = A + (long)(tile_m * 16 + lm) * K;
  int n0 = tile_n * 64 + lm;

  v8f acc0 = {0.f,0.f,0.f,0.f,0.f,0.f,0.f,0.f};
  v8f acc1 = acc0, acc2 = acc0, acc3 = acc0;

  #pragma unroll 2
  for (int k = 0; k < K; k += 4) {
    v2f a = *(const v2f*)(Arow + k + kk);              // 8B contiguous
    const float* Br0 = B + (long)(k + kk) * NC;
    const float* Br1 = Br0 + NC;
    v2f b0, b1, b2, b3;
    b0.x = Br0[n0];      b0.y = Br1[n0];
    b1.x = Br0[n0 + 16]; b1.y = Br1[n0 + 16];
    b2.x = Br0[n0 + 32]; b2.y = Br1[n0 + 32];
    b3.x = Br0[n0 + 48]; b3.y = Br1[n0 + 48];
    acc0 = __builtin_amdgcn_wmma_f32_16x16x4_f32(false, a, false, b0, (short)0, acc0, false, false);
    acc1 = __builtin_amdgcn_wmma_f32_16x16x4_f32(false, a, false, b1, (short)0, acc1, false, false);
    acc2 = __builtin_amdgcn_wmma_f32_16x16x4_f32(false, a, false, b2, (short)0, acc2, false, false);
    acc3 = __builtin_amdgcn_wmma_f32_16x16x4_f32(false, a, false, b3, (short)0, acc3, false, false);
  }

  int rbase = tile_m * 16 + ((lane >> 4) << 3);
  #pragma unroll
  for (int v = 0; v < 8; ++v) {
    float* Crow = C + (long)(rbase + v) * NC;
    Crow[n0]      = acc0[v];
    Crow[n0 + 16] = acc1[v];
    Crow[n0 + 32] = acc2[v];
    Crow[n0 + 48] = acc3[v];
  }
}

// ---------------------------------------------------------------------------
// el[n,h] = sum_d feat[n,h,d]*al[h,d]; er likewise. One thread per (n,h).
// ---------------------------------------------------------------------------
__global__ void eler_kernel(const float* __restrict__ feat,
                            const float* __restrict__ al, const float* __restrict__ ar,
                            float* __restrict__ el, float* __restrict__ er, int Nn) {
  int idx = blockIdx.x * blockDim.x + threadIdx.x;
  if (idx >= Nn * HEADS) return;
  int n = idx >> 2, h = idx & 3;
  const float* f = feat + (long)n * F + h * DH;
  const float* a = al + h * DH;
  const float* b = ar + h * DH;
  float sl = 0.f, sr = 0.f;
  #pragma unroll 8
  for (int d = 0; d < DH; ++d) { float v = f[d]; sl += v * a[d]; sr += v * b[d]; }
  el[idx] = sl; er[idx] = sr;
}

// ---------------------------------------------------------------------------
// Fused per-dst-node edge softmax + weighted aggregation + ELU.
// One 256-thread block per dst node d; thread t owns feature t (head h=t>>6).
// Phase A: per-head segment max (LDS tree over the 64 threads of each head).
// Phase B: per-head sum of exp.
// Phase C: acc_t = sum_j feat[src_j, t] * softmax_j(h)  -- no atomics.
// ELU fused into the final store. transposed=1 emits the final [H][N][D].
// ---------------------------------------------------------------------------
__global__ void gat_edge_fused(const int* __restrict__ src,
                               const int* __restrict__ row_off,
                               const int* __restrict__ eids,
                               const float* __restrict__ el,
                               const float* __restrict__ er,
                               const float* __restrict__ feat,
                               float* __restrict__ out,
                               int Nn, int transposed) {
  __shared__ float red[256];
  int d = blockIdx.x;
  int t = threadIdx.x;
  int h = t >> 6;
  int lane64 = t & 63;
  int start = row_off[d];
  int deg = row_off[d + 1] - start;
  float erd = er[d * HEADS + h];

  // ---- phase A: per-head max over incoming edges ----
  float lm = -INFINITY;
  for (int j = lane64; j < deg; j += 64) {
    int s = src[eids[start + j]];
    float x = el[s * HEADS + h] + erd;
    x = (x > 0.f) ? x : SLOPE * x;
    lm = fmaxf(lm, x);
  }
  red[t] = lm;
  __syncthreads();
  #pragma unroll
  for (int ofs = 32; ofs >= 1; ofs >>= 1) {
    if (lane64 < ofs) red[t] = fmaxf(red[t], red[t + ofs]);
    __syncthreads();
  }
  float mh = red[h << 6];
  __syncthreads();                       // red reused below

  // ---- phase B: per-head sum of exp(x - max) ----
  float ls = 0.f;
  for (int j = lane64; j < deg; j += 64) {
    int s = src[eids[start + j]];
    float x = el[s * HEADS + h] + erd;
    x = (x > 0.f) ? x : SLOPE * x;
    ls += expf(x - mh);
  }
  red[t] = ls;
  __syncthreads();
  #pragma unroll
  for (int ofs = 32; ofs >= 1; ofs >>= 1) {
    if (lane64 < ofs) red[t] += red[t + ofs];
    __syncthreads();
  }
  float inv = (deg > 0) ? 1.f / red[h << 6] : 0.f;

  // ---- phase C: weighted gather, software prefetch of next src row ----
  float acc = 0.f;
  int s_cur = (deg > 0) ? src[eids[start]] : 0;
  for (int j = 0; j < deg; ++j) {
    int s = s_cur;
    if (j + 1 < deg) {
      s_cur = src[eids[start + j + 1]];
      __builtin_prefetch(&feat[(long)s_cur * F + t], 0, 1);  // global_prefetch_b8
    }
    float x = el[s * HEADS + h] + erd;
    x = (x > 0.f) ? x : SLOPE * x;
    float a = expf(x - mh) * inv;
    acc += feat[(long)s * F + t] * a;
  }

  // ---- fused ELU + store ----
  acc = (acc > 0.f) ? acc : (expf(acc) - 1.f);
  if (transposed) out[(long)h * Nn * DH + (long)d * DH + (t & 63)] = acc;
  else            out[(long)d * F + t] = acc;
}

// ---------------------------------------------------------------------------
extern "C" void kernel_launch(void* const* d_in, const int* in_sizes, int n_in,
                              void* d_out, int out_size, void* d_ws, size_t ws_size,
                              hipStream_t stream) {
  const float* x   = (const float*)d_in[0];
  const float* W1  = (const float*)d_in[1];
  const float* al1 = (const float*)d_in[2];
  const float* ar1 = (const float*)d_in[3];
  const float* W2  = (const float*)d_in[4];
  const float* al2 = (const float*)d_in[5];
  const float* ar2 = (const float*)d_in[6];
  const int*   src = (const int*)d_in[7];
  const int*   dst = (const int*)d_in[8];

  const int Nn = in_sizes[0] / F;     // 50000
  const int Ee = in_sizes[7];         // 800000

  // workspace carve-up (4-byte aligned throughout)
  float* ws   = (float*)d_ws;
  float* feat = ws;                               // Nn*F   (reused both layers)
  float* h1   = feat + (long)Nn * F;              // Nn*F
  float* el   = h1   + (long)Nn * F;              // Nn*HEADS
  float* er   = el   + (long)Nn * HEADS;          // Nn*HEADS
  int*   deg  = (int*)(er + (long)Nn * HEADS);    // Nn
  int*   roff = deg + Nn;                         // Nn+1
  int*   cur  = roff + Nn + 1;                    // Nn
  int*   eids = cur + Nn;                         // Ee
  float* outp = (float*)d_out;                    // [H][Nn][DH]

  const dim3 B(256);
  const int gemm_blocks = ((Nn / 16) * (F / 64) * 32 + 255) / 256;
  const int nh_blocks   = (Nn * HEADS + 255) / 256;
  const int e_blocks    = (Ee + 255) / 256;
  const int n_blocks    = (Nn + 255) / 256;

  // -------- CSR build over dst (shared by both layers) --------
  fill_int_kernel<<<n_blocks, B, 0, stream>>>(deg, 0, Nn);
  count_kernel<<<e_blocks, B, 0, stream>>>(dst, deg, Ee);
  scan_kernel<<<1, B, 0, stream>>>(deg, roff, Nn);
  copy_int_kernel<<<n_blocks, B, 0, stream>>>(cur, roff, Nn);
  scatter_kernel<<<e_blocks, B, 0, stream>>>(dst, cur, eids, Ee);

  // -------- layer 1 --------
  gemm256_wmma<<<gemm_blocks, B, 0, stream>>>(x, W1, feat, Nn);
  eler_kernel<<<nh_blocks, B, 0, stream>>>(feat, al1, ar1, el, er, Nn);
  gat_edge_fused<<<Nn, B, 0, stream>>>(src, roff, eids, el, er, feat, h1, Nn, 0);

  // -------- layer 2 --------
  gemm256_wmma<<<gemm_blocks, B, 0, stream>>>(h1, W2, feat, Nn);
  eler_kernel<<<nh_blocks, B, 0, stream>>>(feat, al2, ar2, el, er, Nn);
  gat_edge_fused<<<Nn, B, 0, stream>>>(src, roff, eids, el, er, feat, outp, Nn, 1);
}